// N4_44959717655096
// MI455X (gfx1250) — compile-verified
//
#include <hip/hip_runtime.h>
#include <hip/hip_bf16.h>
#include <stdint.h>

// Problem constants (match reference)
#define NNODES   100000
#define NEDGES   3200000
#define NFEAT    16
#define NLAYERS  4

// Edge-stream kernels: 256 threads stage a 2048-edge chunk into LDS via TDM
#define SCATTER_BLOCK 256
#define CHUNK         2048
#define EDGES_PER_THR (CHUNK / SCATTER_BLOCK)

// Prefix-scan tiling for the 100K-entry degree array
#define TILE   1024
#define NTILES ((NNODES + TILE - 1) / TILE)

typedef unsigned int u32;
typedef unsigned long long u64;
typedef __attribute__((ext_vector_type(4))) u32 u32x4_t;
typedef __attribute__((ext_vector_type(8))) int i32x8_t;
typedef __attribute__((ext_vector_type(4))) int i32x4_t;

#if defined(__gfx1250__) && __has_builtin(__builtin_amdgcn_tensor_load_to_lds) && \
    __has_builtin(__builtin_amdgcn_s_wait_tensorcnt)
#define USE_TDM 1
#else
#define USE_TDM 0
#endif

// addrspace(3) pointers are 32-bit offsets from the workgroup LDS base; the
// generic pointer keeps that offset in its low 32 bits.
__device__ __forceinline__ u32 lds_byte_offset(const void* p) {
  return (u32)(uintptr_t)p;
}

#if USE_TDM
// 1-D TDM descriptor (D# per CDNA5 ISA ch.8): copy tile_elems dwords from
// gbase into LDS at lds_off; tensor_dim0 = tensor_elems so the ragged tail
// zero-fills. 6-arg builtin on this toolchain; groups 2..4 zero for 1-D.
__device__ __forceinline__ void tdm_load_dw_1d(u32 lds_off, const void* gbase,
                                               u32 tile_elems, u32 tensor_elems) {
  const u64 ga = (u64)(uintptr_t)gbase;
  u32x4_t g0;
  g0[0] = 1u;                                              // count=1 (valid)
  g0[1] = lds_off;                                         // lds_addr (bytes)
  g0[2] = (u32)(ga & 0xFFFFFFFFull);                       // global_addr[31:0]
  g0[3] = (u32)((ga >> 32) & 0x01FFFFFFull) | 0x80000000u; // addr[56:32] | type=2
  i32x8_t g1;
  g1[0] = (int)(2u << 16);                                 // no cluster mask, data_size=4B
  g1[1] = (int)((tensor_elems & 0xFFFFu) << 16);           // tensor_dim0[15:0]
  g1[2] = (int)(((tensor_elems >> 16) & 0xFFFFu) | (1u << 16)); // dim0 hi | tensor_dim1=1
  g1[3] = (int)((tile_elems & 0xFFFFu) << 16);             // tile_dim0
  g1[4] = (int)1;                                          // tile_dim1=1, tile_dim2=0
  g1[5] = (int)tensor_elems;                               // tensor_dim0_stride lo
  g1[6] = 0;
  g1[7] = 0;
  i32x4_t z4 = {0, 0, 0, 0};
  i32x8_t z8 = {0, 0, 0, 0, 0, 0, 0, 0};
  __builtin_amdgcn_tensor_load_to_lds(g0, g1, z4, z4, z8, 0);
}
#endif

// ---------------------------------------------------------------------------
// CSR build
// ---------------------------------------------------------------------------
__global__ __launch_bounds__(256)
void gnn_zero_i32(int* __restrict__ p, int n) {
  int i = blockIdx.x * blockDim.x + threadIdx.x;
  if (i < n) p[i] = 0;
}

// pos[e] = slot of edge e within its destination node; deg[d] = in-degree.
__global__ __launch_bounds__(SCATTER_BLOCK)
void gnn_count_kernel(const int* __restrict__ dst, int* __restrict__ pos,
                      int* __restrict__ deg, int nedges) {
  __shared__ int s_dst[CHUNK];
  const int base = blockIdx.x * CHUNK;
  const int tid  = threadIdx.x;
#if USE_TDM
  if (tid == 0) {
    tdm_load_dw_1d(lds_byte_offset(s_dst), dst + base, CHUNK, (u32)(nedges - base));
    __builtin_amdgcn_s_wait_tensorcnt(0);
  }
  __syncthreads();
#else
  for (int i = tid; i < CHUNK; i += SCATTER_BLOCK) {
    const int e = base + i;
    s_dst[i] = (e < nedges) ? dst[e] : 0;
  }
  __syncthreads();
#endif
#pragma unroll
  for (int j = 0; j < EDGES_PER_THR; ++j) {
    const int el = tid + j * SCATTER_BLOCK;
    const int e  = base + el;
    if (e < nedges) pos[e] = atomicAdd(&deg[s_dst[el]], 1);
  }
}

// Tile sums of deg
__global__ __launch_bounds__(256)
void gnn_scan_partial(const int* __restrict__ deg, int* __restrict__ tilesum, int n) {
  __shared__ int sh[256];
  const int t = threadIdx.x, b = blockIdx.x;
  const int i0 = b * TILE + t * 4;
  int s = 0;
#pragma unroll
  for (int k = 0; k < 4; ++k) { int i = i0 + k; if (i < n) s += deg[i]; }
  sh[t] = s; __syncthreads();
  for (int off = 128; off > 0; off >>= 1) {
    if (t < off) sh[t] += sh[t + off];
    __syncthreads();
  }
  if (t == 0) tilesum[b] = sh[0];
}

// In-place exclusive scan of the (small) tile-sum array
__global__ void gnn_scan_tiles(int* __restrict__ tilesum, int ntiles) {
  if (threadIdx.x == 0 && blockIdx.x == 0) {
    int run = 0;
    for (int b = 0; b < ntiles; ++b) { int v = tilesum[b]; tilesum[b] = run; run += v; }
  }
}

// rowstart = exclusive scan of deg (tile-local scan + tile offset)
__global__ __launch_bounds__(256)
void gnn_scan_final(const int* __restrict__ deg, const int* __restrict__ tileoff,
                    int* __restrict__ rowstart, int n) {
  __shared__ int sh[256];
  const int t = threadIdx.x, b = blockIdx.x;
  const int i0 = b * TILE + t * 4;
  int v[4]; int s = 0;
#pragma unroll
  for (int k = 0; k < 4; ++k) { int i = i0 + k; v[k] = (i < n) ? deg[i] : 0; s += v[k]; }
  sh[t] = s; __syncthreads();
  for (int off = 1; off < 256; off <<= 1) {      // Hillis-Steele inclusive
    int x = (t >= off) ? sh[t - off] : 0;
    __syncthreads();
    sh[t] += x;
    __syncthreads();
  }
  int basev = tileoff[b] + sh[t] - s;            // exclusive within tile + tile offset
#pragma unroll
  for (int k = 0; k < 4; ++k) { int i = i0 + k; if (i < n) rowstart[i] = basev; basev += v[k]; }
}

// csr[rowstart[dst[e]] + pos[e]] = {src[e], e}
__global__ __launch_bounds__(SCATTER_BLOCK)
void gnn_fill_kernel(const int* __restrict__ src, const int* __restrict__ dst,
                     const int* __restrict__ pos, const int* __restrict__ rowstart,
                     int2* __restrict__ csr, int nedges) {
  __shared__ int s_src[CHUNK];
  __shared__ int s_dst[CHUNK];
  __shared__ int s_pos[CHUNK];
  const int base = blockIdx.x * CHUNK;
  const int tid  = threadIdx.x;
#if USE_TDM
  if (tid == 0) {
    const u32 rem = (u32)(nedges - base);
    tdm_load_dw_1d(lds_byte_offset(s_src), src + base, CHUNK, rem);
    tdm_load_dw_1d(lds_byte_offset(s_dst), dst + base, CHUNK, rem);
    tdm_load_dw_1d(lds_byte_offset(s_pos), pos + base, CHUNK, rem);
    __builtin_amdgcn_s_wait_tensorcnt(0);
  }
  __syncthreads();
#else
  for (int i = tid; i < CHUNK; i += SCATTER_BLOCK) {
    const int e = base + i;
    const bool ok = e < nedges;
    s_src[i] = ok ? src[e] : 0;
    s_dst[i] = ok ? dst[e] : 0;
    s_pos[i] = ok ? pos[e] : 0;
  }
  __syncthreads();
#endif
#pragma unroll
  for (int j = 0; j < EDGES_PER_THR; ++j) {
    const int el = tid + j * SCATTER_BLOCK;
    const int e  = base + el;
    if (e < nedges) {
      const int slot = rowstart[s_dst[el]] + s_pos[el];
      csr[slot] = make_int2(s_src[el], e);
    }
  }
}

// ---------------------------------------------------------------------------
// Pull layer: 16 threads per node (one per feature), zero float atomics.
// mode 0: hout = sum + h0    mode 1: hout = sigmoid(2*sum)
// ---------------------------------------------------------------------------
__global__ __launch_bounds__(256)
void gnn_pull_kernel(const int* __restrict__ rowstart, const int* __restrict__ deg,
                     const int2* __restrict__ csr, const float* __restrict__ wt,
                     const float* __restrict__ lw, const float* __restrict__ hin,
                     const float* __restrict__ h0, float* __restrict__ hout, int mode) {
  const int t = threadIdx.x;
  const int node = blockIdx.x * 16 + (t >> 4);
  const int f = t & 15;
  if (node >= NNODES) return;
  const int start = rowstart[node];
  const int end   = start + deg[node];
  float acc = 0.f;
  for (int i = start; i < end; ++i) {
    const int2 se = csr[i];                       // {src, edge_id}, one b64 per group
    const float w = wt[se.y] * lw[se.y];
    acc = fmaf(w, hin[(size_t)se.x * NFEAT + f], acc);
  }
  float r;
  if (mode == 0) r = acc + h0[(size_t)node * NFEAT + f];
  else           r = 1.f / (1.f + __expf(-2.f * acc));
  hout[(size_t)node * NFEAT + f] = r;
}

// ---------------------------------------------------------------------------
// Fallback push path (used only if workspace is too small for CSR)
// ---------------------------------------------------------------------------
__global__ __launch_bounds__(256)
void gnn_init_kernel(float* __restrict__ out, const float* __restrict__ h0, int n4) {
  int i = blockIdx.x * blockDim.x + threadIdx.x;
  if (i < n4) {
    float4 v;
    if (h0) v = ((const float4*)h0)[i];
    else    v = make_float4(0.f, 0.f, 0.f, 0.f);
    ((float4*)out)[i] = v;
  }
}

__global__ __launch_bounds__(SCATTER_BLOCK)
void gnn_scatter_kernel(const int* __restrict__ src, const int* __restrict__ dst,
                        const float* __restrict__ wt, const float* __restrict__ lw,
                        const float* __restrict__ hin, float* __restrict__ hout,
                        int nedges) {
  __shared__ int   s_src[CHUNK];
  __shared__ int   s_dst[CHUNK];
  __shared__ float s_wt[CHUNK];
  __shared__ float s_lw[CHUNK];
  const int base = blockIdx.x * CHUNK;
  const int tid  = threadIdx.x;
#if USE_TDM
  if (tid == 0) {
    const u32 rem = (u32)(nedges - base);
    tdm_load_dw_1d(lds_byte_offset(s_src), src + base, CHUNK, rem);
    tdm_load_dw_1d(lds_byte_offset(s_dst), dst + base, CHUNK, rem);
    tdm_load_dw_1d(lds_byte_offset(s_wt),  wt  + base, CHUNK, rem);
    tdm_load_dw_1d(lds_byte_offset(s_lw),  lw  + base, CHUNK, rem);
    __builtin_amdgcn_s_wait_tensorcnt(0);
  }
  __syncthreads();
#else
  for (int i = tid; i < CHUNK; i += SCATTER_BLOCK) {
    const int e = base + i;
    const bool ok = e < nedges;
    s_src[i] = ok ? src[e] : 0;
    s_dst[i] = ok ? dst[e] : 0;
    s_wt[i]  = ok ? wt[e]  : 0.f;
    s_lw[i]  = ok ? lw[e]  : 0.f;
  }
  __syncthreads();
#endif
#pragma unroll
  for (int j = 0; j < EDGES_PER_THR; ++j) {
    const int el = tid + j * SCATTER_BLOCK;
    if (base + el < nedges) {
      const int   sn = s_src[el];
      const int   dn = s_dst[el];
      const float w  = s_wt[el] * s_lw[el];
      const float4* hs = (const float4*)(hin + (size_t)sn * NFEAT);
      const float4 a = hs[0], b = hs[1], c = hs[2], d = hs[3];
      float* o = hout + (size_t)dn * NFEAT;
      atomicAdd(o +  0, w * a.x); atomicAdd(o +  1, w * a.y);
      atomicAdd(o +  2, w * a.z); atomicAdd(o +  3, w * a.w);
      atomicAdd(o +  4, w * b.x); atomicAdd(o +  5, w * b.y);
      atomicAdd(o +  6, w * b.z); atomicAdd(o +  7, w * b.w);
      atomicAdd(o +  8, w * c.x); atomicAdd(o +  9, w * c.y);
      atomicAdd(o + 10, w * c.z); atomicAdd(o + 11, w * c.w);
      atomicAdd(o + 12, w * d.x); atomicAdd(o + 13, w * d.y);
      atomicAdd(o + 14, w * d.z); atomicAdd(o + 15, w * d.w);
    }
  }
}

__global__ __launch_bounds__(256)
void gnn_sigmoid_kernel(const float* __restrict__ hin, float* __restrict__ out, int n4) {
  int i = blockIdx.x * blockDim.x + threadIdx.x;
  if (i < n4) {
    float4 v = ((const float4*)hin)[i];
    float4 r;
    r.x = 1.f / (1.f + __expf(-2.f * v.x));
    r.y = 1.f / (1.f + __expf(-2.f * v.y));
    r.z = 1.f / (1.f + __expf(-2.f * v.z));
    r.w = 1.f / (1.f + __expf(-2.f * v.w));
    ((float4*)out)[i] = r;
  }
}

// ---------------------------------------------------------------------------
extern "C" void kernel_launch(void* const* d_in, const int* in_sizes, int n_in,
                              void* d_out, int out_size, void* d_ws, size_t ws_size,
                              hipStream_t stream) {
  const float* h0  = (const float*)d_in[0];             // [NNODES, NFEAT]
  const int*   ei  = (const int*)d_in[1];               // [2, NEDGES] (int32 per harness)
  const float* wt  = (const float*)d_in[2];             // [NEDGES]
  const float* lwa = (const float*)d_in[3];             // [NLAYERS, NEDGES]
  const int* src = ei;
  const int* dst = ei + NEDGES;
  float* out = (float*)d_out;

  // Workspace layout
  float* hA = (float*)d_ws;                                   // 6.4 MB
  float* hB = hA + (size_t)NNODES * NFEAT;                    // 6.4 MB
  int2*  csr = (int2*)(hB + (size_t)NNODES * NFEAT);          // 25.6 MB
  int*   pos = (int*)(csr + (size_t)NEDGES);                  // 12.8 MB
  int*   deg = pos + (size_t)NEDGES;                          // 0.4 MB
  int*   rowstart = deg + NNODES;                             // 0.4 MB
  int*   tilesum  = rowstart + NNODES;                        // tiny
  const size_t need_csr = (size_t)((char*)(tilesum + NTILES + 16) - (char*)d_ws);

  const dim3 sb(SCATTER_BLOCK), sg((NEDGES + CHUNK - 1) / CHUNK);
  const int n4 = (NNODES * NFEAT) / 4;
  const dim3 eb(256), eg((n4 + 255) / 256);

  if (ws_size >= need_csr) {
    // ---- CSR build (once per call; only 3.2M int atomics) ----
    gnn_zero_i32<<<dim3((NNODES + 255) / 256), dim3(256), 0, stream>>>(deg, NNODES);
    gnn_count_kernel<<<sg, sb, 0, stream>>>(dst, pos, deg, NEDGES);
    gnn_scan_partial<<<dim3(NTILES), dim3(256), 0, stream>>>(deg, tilesum, NNODES);
    gnn_scan_tiles<<<dim3(1), dim3(64), 0, stream>>>(tilesum, NTILES);
    gnn_scan_final<<<dim3(NTILES), dim3(256), 0, stream>>>(deg, tilesum, rowstart, NNODES);
    gnn_fill_kernel<<<sg, sb, 0, stream>>>(src, dst, pos, rowstart, csr, NEDGES);
    // ---- 4 pull layers, residual/sigmoid fused, no float atomics ----
    const dim3 pb(256), pg((NNODES + 15) / 16);
    gnn_pull_kernel<<<pg, pb, 0, stream>>>(rowstart, deg, csr, wt, lwa + 0 * (size_t)NEDGES, h0, h0, hA, 0);
    gnn_pull_kernel<<<pg, pb, 0, stream>>>(rowstart, deg, csr, wt, lwa + 1 * (size_t)NEDGES, hA, h0, hB, 0);
    gnn_pull_kernel<<<pg, pb, 0, stream>>>(rowstart, deg, csr, wt, lwa + 2 * (size_t)NEDGES, hB, h0, hA, 0);
    gnn_pull_kernel<<<pg, pb, 0, stream>>>(rowstart, deg, csr, wt, lwa + 3 * (size_t)NEDGES, hA, h0, out, 1);
  } else {
    // ---- Fallback: atomic push path (needs only 12.8 MB) ----
    gnn_init_kernel<<<eg, eb, 0, stream>>>(hA, h0, n4);
    gnn_scatter_kernel<<<sg, sb, 0, stream>>>(src, dst, wt, lwa + 0 * (size_t)NEDGES, h0, hA, NEDGES);
    gnn_init_kernel<<<eg, eb, 0, stream>>>(hB, h0, n4);
    gnn_scatter_kernel<<<sg, sb, 0, stream>>>(src, dst, wt, lwa + 1 * (size_t)NEDGES, hA, hB, NEDGES);
    gnn_init_kernel<<<eg, eb, 0, stream>>>(hA, h0, n4);
    gnn_scatter_kernel<<<sg, sb, 0, stream>>>(src, dst, wt, lwa + 2 * (size_t)NEDGES, hB, hA, NEDGES);
    gnn_init_kernel<<<eg, eb, 0, stream>>>(hB, nullptr, n4);
    gnn_scatter_kernel<<<sg, sb, 0, stream>>>(src, dst, wt, lwa + 3 * (size_t)NEDGES, hA, hB, NEDGES);
    gnn_sigmoid_kernel<<<eg, eb, 0, stream>>>(hB, out, n4);
  }
}